// EdgeGCN_71597104824953
// MI455X (gfx1250) — compile-verified
//
#include <hip/hip_runtime.h>
#include <hip/hip_bf16.h>

typedef __attribute__((ext_vector_type(2))) float v2f;
typedef __attribute__((ext_vector_type(8))) float v8f;

// ---------------------------------------------------------------- utilities

__global__ void eg_fill_ones(float* __restrict__ p, int n) {
    int i = blockIdx.x * blockDim.x + threadIdx.x;
    if (i < n) p[i] = 1.0f;
}

__global__ void eg_deg_accum(const int* __restrict__ dst, float* __restrict__ deg, int e) {
    int i = blockIdx.x * blockDim.x + threadIdx.x;
    if (i < e) atomicAdd(&deg[dst[i]], 1.0f);   // exact: integer-valued floats
}

__global__ void eg_rsqrt_ip(float* __restrict__ p, int n) {
    int i = blockIdx.x * blockDim.x + threadIdx.x;
    if (i < n) p[i] = rsqrtf(p[i]);             // deg >= 1 always (self loop)
}

// ------------------------------------------------- WMMA GEMM: [n,K]x[K,16]
// One wave computes one 16-row tile via V_WMMA_F32_16X16X4_F32.
// A (16x4 f32): lanes 0-15 M=lane, VGPR{0,1}=K{0,1}; lanes 16-31 VGPR{0,1}=K{2,3}.
// B (4x16 f32): VGPR0 lanes0-15 K=0 / lanes16-31 K=2; VGPR1 K=1 / K=3.
// C/D (16x16 f32): VGPR v -> M=v (lanes 0-15) or M=v+8 (lanes 16-31), N=lane&15.
// Epilogue writes Out = A@W and Agg = (A@W)*dinv^2 (GCN self-loop term).
template <int K>
__global__ void eg_gemm_n16_wmma(const float* __restrict__ A,
                                 const float* __restrict__ W,     // K x 16 row-major
                                 float* __restrict__ Out,         // n x 16
                                 const float* __restrict__ dinv,  // n
                                 float* __restrict__ Agg,         // n x 16
                                 int nrows) {
    __shared__ float sW[K * 16];
    for (int t = threadIdx.x; t < K * 16; t += blockDim.x) sW[t] = W[t];
    __syncthreads();

    const int wave   = threadIdx.x >> 5;
    const int tile   = blockIdx.x * (blockDim.x >> 5) + wave;
    const int ntiles = (nrows + 15) >> 4;
    if (tile >= ntiles) return;                 // uniform per-wave exit

    const int lane = threadIdx.x & 31;
    const int half = lane >> 4;                 // 0: K pair {0,1}   1: {2,3}
    const int m    = lane & 15;                 // A row in tile / B,C column
    int row = tile * 16 + m;
    row = (row < nrows) ? row : (nrows - 1);    // clamp: unconditional loads
    const float* arow = A + (size_t)row * K;

    v8f c = {};
#pragma unroll
    for (int k0 = 0; k0 < K; k0 += 4) {
        const int ka = k0 + half * 2;
        v2f a, b;
        a[0] = arow[ka];
        a[1] = arow[ka + 1];
        b[0] = sW[ka * 16 + m];
        b[1] = sW[(ka + 1) * 16 + m];
        c = __builtin_amdgcn_wmma_f32_16x16x4_f32(
                /*neg_a=*/false, a, /*neg_b=*/false, b,
                /*c_mod=*/(short)0, c, /*reuse_a=*/false, /*reuse_b=*/false);
    }

    const int rbase = tile * 16 + half * 8;
#pragma unroll
    for (int v = 0; v < 8; ++v) {
        const int r = rbase + v;
        if (r < nrows) {
            const size_t o = (size_t)r * 16 + m;
            float di = dinv[r];
            Out[o] = c[v];
            Agg[o] = c[v] * di * di;
        }
    }
}

// --------------------------------------------------------- edge aggregation

// one thread per (edge, feature); lane low 4 bits = feature -> coalesced
// gathers and 64B-contiguous atomic groups per destination row
__global__ void eg_edge_scatter(const float* __restrict__ xw, const int* __restrict__ src,
                                const int* __restrict__ dst, const float* __restrict__ dinv,
                                float* __restrict__ agg, int e) {
    long long i = (long long)blockIdx.x * blockDim.x + threadIdx.x;
    if (i >= (long long)e * 16) return;
    int ed = (int)(i >> 4);
    int j  = (int)(i & 15);
    int s = src[ed], d = dst[ed];
    float norm = dinv[s] * dinv[d];
    atomicAdd(&agg[(size_t)d * 16 + j], xw[(size_t)s * 16 + j] * norm);
}

__global__ void eg_bias_act(float* __restrict__ h, const float* __restrict__ b,
                            int n, int do_relu) {
    int i = blockIdx.x * blockDim.x + threadIdx.x;
    if (i >= n * 16) return;
    float v = h[i] + b[i & 15];
    h[i] = do_relu ? fmaxf(v, 0.0f) : v;
}

// ------------------------------------------------------------ edge predictor
// out[e] = b + sum_k h[src][k]*Wfc[k] + h[dst][k]*Wfc[16+k]
__global__ void eg_edge_pred(const float* __restrict__ h, const int* __restrict__ src,
                             const int* __restrict__ dst, const float* __restrict__ wfc,
                             const float* __restrict__ bfc, float* __restrict__ out, int e) {
    int i = blockIdx.x * blockDim.x + threadIdx.x;
    if (i >= e) return;
    int s = src[i], d = dst[i];
    const float4* hs = (const float4*)(h + (size_t)s * 16);
    const float4* hd = (const float4*)(h + (size_t)d * 16);
    const float4* wa = (const float4*)wfc;        // uniform -> scalar loads
    const float4* wb = (const float4*)(wfc + 16);
    float acc = bfc[0];
#pragma unroll
    for (int q = 0; q < 4; ++q) {
        float4 a = hs[q], x = wa[q];
        float4 b = hd[q], y = wb[q];
        acc += a.x * x.x + a.y * x.y + a.z * x.z + a.w * x.w;
        acc += b.x * y.x + b.y * y.y + b.z * y.z + b.w * y.w;
    }
    out[i] = acc;
}

// ------------------------------------------------------------------ launch

extern "C" void kernel_launch(void* const* d_in, const int* in_sizes, int n_in,
                              void* d_out, int out_size, void* d_ws, size_t ws_size,
                              hipStream_t stream) {
    const float* x   = (const float*)d_in[0];
    const int*   ei  = (const int*)d_in[1];
    const float* W1  = (const float*)d_in[2];
    const float* b1  = (const float*)d_in[3];
    const float* W2  = (const float*)d_in[4];
    const float* b2  = (const float*)d_in[5];
    const float* Wfc = (const float*)d_in[6];
    const float* bfc = (const float*)d_in[7];
    float*       out = (float*)d_out;

    const int F_IN = 128;
    const int N = in_sizes[0] / F_IN;       // 100000
    const int E = in_sizes[1] / 2;          // 3200000
    const int* src = ei;
    const int* dst = ei + E;

    // workspace layout (floats): dinv[N] | P[N*16] | Q[N*16] | R[N*16]
    float* dinv = (float*)d_ws;
    float* P    = dinv + (size_t)N;         // N*4 = 400000 B -> 64B aligned
    float* Q    = P + (size_t)N * 16;
    float* R    = Q + (size_t)N * 16;

    const int B = 256;
    const int gN   = (N + B - 1) / B;
    const int gE   = (E + B - 1) / B;
    const int gNF  = (N * 16 + B - 1) / B;
    const long long ef = (long long)E * 16;
    const int gEF  = (int)((ef + B - 1) / B);
    const int ntiles = (N + 15) / 16;
    const int gT   = (ntiles + (B / 32) - 1) / (B / 32);

    // degree -> dinv
    eg_fill_ones<<<gN, B, 0, stream>>>(dinv, N);
    eg_deg_accum<<<gE, B, 0, stream>>>(dst, dinv, E);
    eg_rsqrt_ip<<<gN, B, 0, stream>>>(dinv, N);

    // layer 1: P = x@W1, Q = P*dinv^2 ; Q += scatter ; Q = relu(Q + b1)
    eg_gemm_n16_wmma<128><<<gT, B, 0, stream>>>(x, W1, P, dinv, Q, N);
    eg_edge_scatter<<<gEF, B, 0, stream>>>(P, src, dst, dinv, Q, E);
    eg_bias_act<<<gNF, B, 0, stream>>>(Q, b1, N, 1);

    // layer 2: P = Q@W2, R = P*dinv^2 ; R += scatter ; R = R + b2
    eg_gemm_n16_wmma<16><<<gT, B, 0, stream>>>(Q, W2, P, dinv, R, N);
    eg_edge_scatter<<<gEF, B, 0, stream>>>(P, src, dst, dinv, R, E);
    eg_bias_act<<<gNF, B, 0, stream>>>(R, b2, N, 0);

    // edge prediction
    eg_edge_pred<<<gE, B, 0, stream>>>(R, src, dst, Wfc, bfc, out, E);
}